// TDRLifting_15788299780583
// MI455X (gfx1250) — compile-verified
//
#include <hip/hip_runtime.h>
#include <hip/hip_bf16.h>
#include <math.h>

typedef __attribute__((ext_vector_type(2))) float v2f;
typedef __attribute__((ext_vector_type(8))) float v8f;

#define B_      8
#define NC_     6
#define N_      900
#define D_      48
#define M_      (NC_*N_*D_)        // 259200 per batch
#define MAXQ_   400
#define NTILE_  ((N_+7)/8)         // 113 n-tiles of 8 waves
#define NCHUNK_ ((M_+255)/256)     // 1013
#define NBIN_   4096
#define IMG_AREA_HI_ (800.0f*448.0f*0.55f)   // 197120
#define SR_MIN_  (-51.2f)
#define SR_RANGE_ (102.4f)
#define BIN_SC_  ((float)NBIN_/IMG_AREA_HI_)

__device__ __forceinline__ float nan2num(float x){
  if (isnan(x)) return 0.0f;
  if (isinf(x)) return x > 0.0f ? 1e6f : -1e6f;
  return x;
}

__device__ __forceinline__ void inv3(const float a[3][3], float o[3][3]){
  float c00 =  a[1][1]*a[2][2]-a[1][2]*a[2][1];
  float c01 = -(a[1][0]*a[2][2]-a[1][2]*a[2][0]);
  float c02 =  a[1][0]*a[2][1]-a[1][1]*a[2][0];
  float det = a[0][0]*c00 + a[0][1]*c01 + a[0][2]*c02;
  float id  = 1.0f/det;
  o[0][0]=c00*id;
  o[0][1]=-(a[0][1]*a[2][2]-a[0][2]*a[2][1])*id;
  o[0][2]= (a[0][1]*a[1][2]-a[0][2]*a[1][1])*id;
  o[1][0]=c01*id;
  o[1][1]= (a[0][0]*a[2][2]-a[0][2]*a[2][0])*id;
  o[1][2]=-(a[0][0]*a[1][2]-a[0][2]*a[1][0])*id;
  o[2][0]=c02*id;
  o[2][1]=-(a[0][0]*a[2][1]-a[0][1]*a[2][0])*id;
  o[2][2]= (a[0][0]*a[1][1]-a[0][1]*a[1][0])*id;
  for (int r=0;r<3;r++) for (int c=0;c<3;c++) o[r][c]=nan2num(o[r][c]);
}

// --- per-(b,c) fused matrices: rows 0-3 = T = E*Kinv4, 4-7 = M2 = Einv*T,
//     rows 8-10 = M3 = K3*M2[:3], rows 11-15 zero.  64 floats per (b,c).
__global__ void k_mats(const float* __restrict__ K, const float* __restrict__ E,
                       float* __restrict__ mats){
  int i = blockIdx.x*blockDim.x + threadIdx.x;
  if (i >= B_*NC_) return;
  const float* Kp = K + (size_t)i*16;
  const float* Ep = E + (size_t)i*16;
  float K3[3][3], Ki[3][3];
  for (int r=0;r<3;r++) for (int c=0;c<3;c++) K3[r][c]=Kp[r*4+c];
  inv3(K3, Ki);
  float K4[4][4] = {{Ki[0][0],Ki[0][1],Ki[0][2],0.f},
                    {Ki[1][0],Ki[1][1],Ki[1][2],0.f},
                    {Ki[2][0],Ki[2][1],Ki[2][2],0.f},
                    {0.f,0.f,0.f,1.f}};
  float E4[4][4];
  for (int r=0;r<4;r++) for (int c=0;c<4;c++) E4[r][c]=Ep[r*4+c];
  float T[4][4];
  for (int r=0;r<4;r++) for (int c=0;c<4;c++){
    float s=0.f; for (int k=0;k<4;k++) s += E4[r][k]*K4[k][c]; T[r][c]=s;
  }
  // affine inverse of E (E row3 is (0,0,0,1) by construction)
  float R[3][3], Ri[3][3];
  for (int r=0;r<3;r++) for (int c=0;c<3;c++) R[r][c]=E4[r][c];
  inv3(R, Ri);
  float Ei[4][4];
  for (int r=0;r<3;r++){
    for (int c=0;c<3;c++) Ei[r][c]=Ri[r][c];
    float s=0.f; for (int k=0;k<3;k++) s += Ri[r][k]*E4[k][3];
    Ei[r][3]=nan2num(-s);
  }
  Ei[3][0]=0.f; Ei[3][1]=0.f; Ei[3][2]=0.f; Ei[3][3]=1.f;
  float M2[4][4];
  for (int r=0;r<4;r++) for (int c=0;c<4;c++){
    float s=0.f; for (int k=0;k<4;k++) s += Ei[r][k]*T[k][c]; M2[r][c]=s;
  }
  float M3[3][4];
  for (int r=0;r<3;r++) for (int c=0;c<4;c++){
    float s=0.f; for (int k=0;k<3;k++) s += K3[r][k]*M2[k][c]; M3[r][c]=s;
  }
  float* o = mats + (size_t)i*64;
  for (int r=0;r<4;r++) for (int c=0;c<4;c++) o[r*4+c]       = T[r][c];
  for (int r=0;r<4;r++) for (int c=0;c<4;c++) o[(4+r)*4+c]   = M2[r][c];
  for (int r=0;r<3;r++) for (int c=0;c<4;c++) o[(8+r)*4+c]   = M3[r][c];
  for (int j=44;j<64;j++) o[j]=0.f;
}

// --- main projection kernel: 8 waves/block, one n per wave, WMMA over 16 depths.
//     Also builds the per-batch area histogram (fused; hist zeroed by k_init).
__global__ void __launch_bounds__(256)
k_main(const float* __restrict__ boxes, const float* __restrict__ mats,
       float* __restrict__ ptsX, float* __restrict__ ptsY, float* __restrict__ ptsZ,
       float* __restrict__ key, int* __restrict__ hist){
  __shared__ float sM[64];
  int bx   = blockIdx.x;
  int tile = bx % NTILE_;
  int c    = (bx / NTILE_) % NC_;
  int b    = bx / (NTILE_*NC_);
  if (threadIdx.x < 64) sM[threadIdx.x] = mats[(size_t)(b*NC_+c)*64 + threadIdx.x];
  __syncthreads();
  int wv = threadIdx.x >> 5;
  int lane = threadIdx.x & 31;
  int n = tile*8 + wv;
  if (n >= N_) return;   // wave-uniform exit; no barriers after this point
  const float* bp = boxes + ((size_t)(b*NC_+c)*N_ + n)*4;
  float x1=bp[0], y1=bp[1], x2=bp[2], y2=bp[3];
  float cx = 0.5f*(x1+x2), cy = 0.5f*(y1+y2);
  float area = (x2-x1)*(y2-y1);
  bool  valid = (area > 5.0f) && (area < IMG_AREA_HI_);
  int   abin  = min(NBIN_-1, (int)(area*BIN_SC_));
  int  row = lane & 15;
  bool lo  = lane < 16;
  // A-matrix 16x4 f32 layout: lanes0-15 hold K=0,1; lanes16-31 hold K=2,3
  v2f A;
  A.x = sM[row*4 + (lo ? 0 : 2)];
  A.y = sM[row*4 + (lo ? 1 : 3)];
  for (int t=0;t<3;t++){
    int d_idx = t*16 + row;
    float depth = 1.0f + (float)d_idx * (59.0f/47.0f);   // linspace(1,60,48)
    // B-matrix 4x16: column = u-vector of point d_idx; lanes0-15 rows 0,1; 16-31 rows 2,3
    v2f Bv;
    Bv.x = lo ? cx*depth : depth;
    Bv.y = lo ? cy*depth : 1.0f;
    v8f Cz = {0.f,0.f,0.f,0.f,0.f,0.f,0.f,0.f};
    v8f Dm = __builtin_amdgcn_wmma_f32_16x16x4_f32(
        false, A, false, Bv, (short)0, Cz, false, false);
    // rows 8-10 (q = K3*pch.xyz) live in lanes 16-31; fetch to low half
    float qx = __shfl(Dm[0], row+16, 32);
    float qy = __shfl(Dm[1], row+16, 32);
    float qz = __shfl(Dm[2], row+16, 32);
    if (lo){
      float p3x=Dm[0], p3y=Dm[1], p3z=Dm[2];          // points_3d
      float pchz=Dm[6], pchw=Dm[7];
      float iw   = __builtin_amdgcn_rcpf(pchw + 1e-6f);
      float dsc  = pchz*iw;                            // depth_scale
      float pihx = qx*iw, pihy = qy*iw, pihz = qz*iw;
      float rz   = __builtin_amdgcn_rcpf(pihz + 1e-6f);
      float imx = pihx*rz, imy = pihy*rz;
      float bs = 400.0f*__builtin_amdgcn_rcpf(dsc + 1e-6f);
      bs = fminf(fmaxf(bs, 8.0f), 200.0f);
      float h = 0.5f*bs;
      float px1=imx-h, py1=imy-h, px2=imx+h, py2=imy+h;
      float ix1=fmaxf(px1,x1), iy1=fmaxf(py1,y1);
      float ix2=fminf(px2,x2), iy2=fminf(py2,y2);
      float inter = fmaxf(ix2-ix1,0.f)*fmaxf(iy2-iy1,0.f);
      float iou = inter*__builtin_amdgcn_rcpf(bs*bs + area - inter + 1e-6f);
      bool  msk = (iou > 0.05f) && valid;
      int m = (c*N_ + n)*D_ + d_idx;
      size_t gi = (size_t)b*M_ + m;
      ptsX[gi]=p3x; ptsY[gi]=p3y; ptsZ[gi]=p3z;
      key[gi] = msk ? area : -__builtin_huge_valf();
      if (msk) atomicAdd(&hist[b*NBIN_ + abin], 1);
    }
  }
}

// --- init outputs to padded state + zero histograms
__global__ void k_init(float* __restrict__ out, int* __restrict__ hist){
  int i = blockIdx.x*blockDim.x + threadIdx.x;
  float padref = fminf(fmaxf((0.0f - SR_MIN_)/(SR_RANGE_+1e-6f),0.f),1.f);
  if (i < B_*MAXQ_*3) out[i] = padref;
  if (i < B_*MAXQ_)   out[B_*MAXQ_*3 + i] = 1.0f;
  if (i < B_*NBIN_)   hist[i] = 0;
}

// --- threshold search: 16 bins/thread, block scan from the top, 16-bin tail.
__global__ void k_thresh(const int* __restrict__ hist, int* __restrict__ binfo){
  int b = blockIdx.x, tid = threadIdx.x;   // 256 threads
  __shared__ int buf[256];
  __shared__ int sFound[2];
  const int* h = hist + b*NBIN_;
  int s = 0;
  #pragma unroll
  for (int j=0;j<16;j++) s += h[(255-tid)*16 + j];   // reversed segment sums
  buf[tid] = s;
  if (tid == 0){ sFound[0] = -1; sFound[1] = 0; }
  __syncthreads();
  for (int off=1; off<256; off<<=1){
    int t = (tid>=off) ? buf[tid-off] : 0;
    __syncthreads();
    buf[tid] += t;
    __syncthreads();
  }
  int total = buf[255];
  int incl  = buf[tid];
  int prev  = (tid==0) ? 0 : buf[tid-1];
  if (incl >= MAXQ_ && prev < MAXQ_){
    sFound[0] = 255 - tid;   // segment containing the 400-crossing
    sFound[1] = prev;        // count strictly above this segment
  }
  __syncthreads();
  if (tid == 0){
    int T=-1, nAb=0;
    int sidx = sFound[0];
    if (sidx >= 0){
      int cum = sFound[1];
      for (int bin = sidx*16+15; bin >= sidx*16; --bin){
        int c2 = cum + h[bin];
        if (c2 >= MAXQ_){ T=bin; nAb=cum; break; }
        cum = c2;
      }
    }
    binfo[b*4+0]=total; binfo[b*4+1]=T; binfo[b*4+2]=nAb; binfo[b*4+3]=0;
  }
}

__global__ void k_chunk(const float* __restrict__ key, const int* __restrict__ binfo,
                        int* __restrict__ cA, int* __restrict__ cB, int* __restrict__ cE){
  int b = blockIdx.y, chunk = blockIdx.x;
  int e = chunk*256 + threadIdx.x;
  __shared__ int s[3];
  if (threadIdx.x < 3) s[threadIdx.x] = 0;
  __syncthreads();
  int T = binfo[b*4+1];
  if (e < M_){
    float kv = key[(size_t)b*M_ + e];
    if (isfinite(kv)){
      atomicAdd(&s[0],1);
      int bin = min(NBIN_-1, (int)(kv*BIN_SC_));
      if (bin > T) atomicAdd(&s[1],1);
      else if (bin == T) atomicAdd(&s[2],1);
    }
  }
  __syncthreads();
  if (threadIdx.x == 0){
    cA[b*NCHUNK_+chunk]=s[0]; cB[b*NCHUNK_+chunk]=s[1]; cE[b*NCHUNK_+chunk]=s[2];
  }
}

__global__ void k_prefix(const int* cA,const int* cB,const int* cE,
                         int* pA,int* pB,int* pE){
  int b = blockIdx.x, tid = threadIdx.x;
  __shared__ int buf[256];
  const int* cs[3] = {cA,cB,cE};
  int*       ps[3] = {pA,pB,pE};
  for (int a=0;a<3;a++){
    const int* c = cs[a] + b*NCHUNK_;
    int*       p = ps[a] + b*NCHUNK_;
    int carry = 0;
    for (int base=0; base<NCHUNK_; base+=256){
      int idx = base + tid;
      int v = (idx < NCHUNK_) ? c[idx] : 0;
      buf[tid] = v; __syncthreads();
      for (int off=1; off<256; off<<=1){
        int t = (tid>=off) ? buf[tid-off] : 0;
        __syncthreads();
        buf[tid] += t;
        __syncthreads();
      }
      if (idx < NCHUNK_) p[idx] = buf[tid] - v + carry;
      int btot = buf[255];
      __syncthreads();
      carry += btot;
    }
    __syncthreads();
  }
}

__device__ __forceinline__ int scan256(int flag, int tid, int* sWave){
  __syncthreads();
  unsigned long long bal = __ballot(flag);
  unsigned int b32 = (unsigned int)bal;           // wave32: low 32 bits
  int lane = tid & 31, wv = tid >> 5;
  int pre = __popc(b32 & ((1u<<lane)-1u));
  if (lane == 0) sWave[wv] = __popc(b32);
  __syncthreads();
  int wpre = 0;
  for (int i=0;i<wv;i++) wpre += sWave[i];
  return wpre + pre;
}

__global__ void k_select(const float* __restrict__ key,
                         const float* __restrict__ ptsX, const float* __restrict__ ptsY,
                         const float* __restrict__ ptsZ,
                         const int* __restrict__ binfo,
                         const int* __restrict__ pA, const int* __restrict__ pB,
                         const int* __restrict__ pE, float* __restrict__ out){
  int b = blockIdx.y, chunk = blockIdx.x, tid = threadIdx.x;
  int e = chunk*256 + tid;
  __shared__ int sWave[8];
  int total = binfo[b*4+0], T = binfo[b*4+1], nAb = binfo[b*4+2];
  float kv = (e < M_) ? key[(size_t)b*M_ + e] : -__builtin_huge_valf();
  int fin = isfinite(kv) ? 1 : 0;
  int bin = fin ? min(NBIN_-1, (int)(kv*BIN_SC_)) : -2;
  int fAb = (fin && bin >  T) ? 1 : 0;
  int fEq = (fin && bin == T) ? 1 : 0;
  int lA  = scan256(fin, tid, sWave);
  int lB  = scan256(fAb, tid, sWave);
  int lE  = scan256(fEq, tid, sWave);
  int slot = -1;
  if (total <= MAXQ_){
    if (fin) slot = pA[b*NCHUNK_+chunk] + lA;
  } else {
    if (fAb) slot = pB[b*NCHUNK_+chunk] + lB;
    else if (fEq){
      int s = nAb + pE[b*NCHUNK_+chunk] + lE;
      if (s < MAXQ_) slot = s;
    }
  }
  if (slot >= 0 && slot < MAXQ_){
    size_t gi = (size_t)b*M_ + e;
    const float inv = 1.0f/(SR_RANGE_+1e-6f);
    float r0 = fminf(fmaxf((ptsX[gi]-SR_MIN_)*inv,0.f),1.f);
    float r1 = fminf(fmaxf((ptsY[gi]-SR_MIN_)*inv,0.f),1.f);
    float r2 = fminf(fmaxf((ptsZ[gi]-SR_MIN_)*inv,0.f),1.f);
    float* ro = out + ((size_t)b*MAXQ_ + slot)*3;
    ro[0]=r0; ro[1]=r1; ro[2]=r2;
    out[B_*MAXQ_*3 + b*MAXQ_ + slot] = 0.0f;
  }
}

extern "C" void kernel_launch(void* const* d_in, const int* in_sizes, int n_in,
                              void* d_out, int out_size, void* d_ws, size_t ws_size,
                              hipStream_t stream) {
  const float* boxes = (const float*)d_in[0];
  const float* K     = (const float*)d_in[1];
  const float* E     = (const float*)d_in[2];
  float* out = (float*)d_out;
  // workspace carve (~33.5 MB)
  float* ptsX = (float*)d_ws;
  float* ptsY = ptsX + (size_t)B_*M_;
  float* ptsZ = ptsY + (size_t)B_*M_;
  float* key  = ptsZ + (size_t)B_*M_;
  float* mats = key  + (size_t)B_*M_;
  int* hist  = (int*)(mats + (size_t)B_*NC_*64);
  int* binfo = hist + B_*NBIN_;
  int* cA = binfo + B_*4;
  int* cB = cA + B_*NCHUNK_;
  int* cE = cB + B_*NCHUNK_;
  int* pA = cE + B_*NCHUNK_;
  int* pB = pA + B_*NCHUNK_;
  int* pE = pB + B_*NCHUNK_;

  k_init  <<<128, 256, 0, stream>>>(out, hist);
  k_mats  <<<1, 64, 0, stream>>>(K, E, mats);
  k_main  <<<B_*NC_*NTILE_, 256, 0, stream>>>(boxes, mats, ptsX, ptsY, ptsZ, key, hist);
  k_thresh<<<B_, 256, 0, stream>>>(hist, binfo);
  k_chunk <<<dim3(NCHUNK_, B_), 256, 0, stream>>>(key, binfo, cA, cB, cE);
  k_prefix<<<B_, 256, 0, stream>>>(cA, cB, cE, pA, pB, pE);
  k_select<<<dim3(NCHUNK_, B_), 256, 0, stream>>>(key, ptsX, ptsY, ptsZ, binfo, pA, pB, pE, out);
}